// QLayerNorm_15470472200509
// MI455X (gfx1250) — compile-verified
//
#include <hip/hip_runtime.h>
#include <stdint.h>

// ---------------------------------------------------------------------------
// QLayerNorm for MI455X (gfx1250): B=4, S=8192, D=1024, fp32.
// Bandwidth-bound (256 MB @ 23.3 TB/s ~= 11 us floor). Strategy:
//   - 8 rows per 256-thread block, staged into LDS with one TDM
//     tensor_load_to_lds (32 KB tile), waited with s_wait_tensorcnt.
//   - one wave32 per row: ds_load_b128 reads, shfl_xor reductions,
//     literal 4-step Newton recurrence, fused scale+bias, NT b128 stores.
// ---------------------------------------------------------------------------

typedef unsigned int v4u __attribute__((ext_vector_type(4)));
typedef int          v8i __attribute__((ext_vector_type(8)));
typedef int          v4i __attribute__((ext_vector_type(4)));
typedef float        v4f __attribute__((ext_vector_type(4)));   // native 128b vector

#define LN_D            1024
#define ROWS_PER_BLOCK  8
#define LN_EPS          5e-05f

__global__ __launch_bounds__(256) void qlayernorm_gfx1250_kernel(
    const float* __restrict__ x,
    const float* __restrict__ weight,
    const float* __restrict__ bias,
    float* __restrict__ out,
    int nrows)
{
    // 8 rows x 1024 f32 = 32 KB. Sole static LDS object -> assumed offset 0.
    __shared__ float smem[ROWS_PER_BLOCK * LN_D];

    const int tid  = threadIdx.x;
    const int wave = tid >> 5;
    const int lane = tid & 31;
    const int row0 = blockIdx.x * ROWS_PER_BLOCK;

    // Warm weight/bias (8 KB, reused by every block) - keep them cache-resident.
    __builtin_prefetch(weight + lane * 32, 0, 3);
    __builtin_prefetch(bias   + lane * 32, 0, 3);

    // The compiler cannot see that the TDM writes `smem`; without a visible
    // (potential) store it folds all LDS loads to undef and deletes the body.
    // Opaque never-taken store marks smem as may-written at zero runtime cost.
    int zero = 0;
    asm volatile("" : "+s"(zero));
    if (zero) smem[tid] = 0.0f;

    // ---- TDM: wave 0 pulls the 8x1024 f32 tile into LDS -------------------
    if (tid < 32) {
        const uint64_t gaddr =
            (uint64_t)(uintptr_t)(x + (size_t)row0 * LN_D);

        // D# group 0 (128b): count=1, lds_addr=0, global_addr, type=2.
        v4u g0;
        g0.x = 1u;                                      // count = 1 (valid)
        g0.y = 0u;                                      // lds_addr = 0
        g0.z = (uint32_t)(gaddr & 0xFFFFFFFFu);         // global_addr[31:0]
        g0.w = (uint32_t)((gaddr >> 32) & 0x01FFFFFFu)  // global_addr[56:32]
             | (2u << 30);                              // type = 2 ("image")

        // D# group 1 (256b).
        v8i g1;
        g1[0] = (int)(2u << 16);                        // data_size = 4 bytes
        g1[1] = (int)(((uint32_t)LN_D & 0xFFFFu) << 16);        // tensor_dim0[15:0]
        g1[2] = (int)(((uint32_t)nrows & 0xFFFFu) << 16);       // dim0 hi=0 | tensor_dim1[15:0]
        g1[3] = (int)(((uint32_t)nrows >> 16) & 0xFFFFu)        // tensor_dim1[31:16]
              | (int)(((uint32_t)LN_D & 0xFFFFu) << 16);        // tile_dim0 = 1024
        g1[4] = ROWS_PER_BLOCK;                         // tile_dim1 = 8, tile_dim2 = 0
        g1[5] = LN_D;                                   // tensor_dim0_stride = 1024
        g1[6] = 0;                                      // stride hi / dim1_stride lo
        g1[7] = 0;

        v4i gz4 = {0, 0, 0, 0};                         // groups 2/3 unused (2D tile)
        v8i gz8 = {0, 0, 0, 0, 0, 0, 0, 0};             // extra group (clang-23 6-arg form)

        __builtin_amdgcn_tensor_load_to_lds(g0, g1, gz4, gz4, gz8, 0);
        __builtin_amdgcn_s_wait_tensorcnt(0);           // data landed in LDS
    }
    __syncthreads();                                    // release the other 7 waves
    asm volatile("" ::: "memory");                      // don't hoist LDS reads

    // ---- one wave per row --------------------------------------------------
    const v4f* rowp = (const v4f*)(smem + wave * LN_D);

    v4f v[8];
    float sum = 0.0f;
    #pragma unroll
    for (int j = 0; j < 8; ++j) {
        v[j] = rowp[j * 32 + lane];                     // ds_load_b128
        sum += v[j].x + v[j].y + v[j].z + v[j].w;
    }
    #pragma unroll
    for (int o = 16; o >= 1; o >>= 1) sum += __shfl_xor(sum, o, 32);
    const float mean = sum * (1.0f / (float)LN_D);

    float sq = 0.0f;
    #pragma unroll
    for (int j = 0; j < 8; ++j) {
        v[j] -= mean;                                   // vector splat subtract
        sq += v[j].x * v[j].x + v[j].y * v[j].y
            + v[j].z * v[j].z + v[j].w * v[j].w;
    }
    #pragma unroll
    for (int o = 16; o >= 1; o >>= 1) sq += __shfl_xor(sq, o, 32);
    const float var = sq * (1.0f / (float)LN_D);

    // Faithful 4-step recurrence: s_{n+1} = (s_n / a + a) * 0.5, s_0 = a.
    const float a = var + LN_EPS;
    float s = (a / a + a) * 0.5f;
    s = (s / a + a) * 0.5f;
    s = (s / a + a) * 0.5f;
    s = (s / a + a) * 0.5f;
    const float inv = 1.0f / s;

    const int row = row0 + wave;
    v4f* __restrict__ orow = (v4f*)(out + (size_t)row * LN_D);
    const v4f* wv = (const v4f*)weight;
    const v4f* bv = (const v4f*)bias;

    #pragma unroll
    for (int j = 0; j < 8; ++j) {
        const int c = j * 32 + lane;
        const v4f wgt = wv[c];
        const v4f bs  = bv[c];
        const v4f r   = v[j] * inv * wgt + bs;
        __builtin_nontemporal_store(r, &orow[c]);       // NT global_store_b128
    }
}

extern "C" void kernel_launch(void* const* d_in, const int* in_sizes, int n_in,
                              void* d_out, int out_size, void* d_ws, size_t ws_size,
                              hipStream_t stream) {
    const float* x      = (const float*)d_in[0];
    const float* weight = (const float*)d_in[1];
    const float* bias   = (const float*)d_in[2];
    float*       out    = (float*)d_out;

    const int nrows = in_sizes[0] / LN_D;               // 32768
    const int grid  = nrows / ROWS_PER_BLOCK;           // 4096 blocks

    qlayernorm_gfx1250_kernel<<<grid, 256, 0, stream>>>(x, weight, bias, out, nrows);
}